// PathRaster2d_67499706024491
// MI455X (gfx1250) — compile-verified
//
#include <hip/hip_runtime.h>

// PathRaster2d on gfx1250 (MI455X, wave32).
//
// Roofline: 151 MB of fp32 output, ~0.5 GFLOP of math -> store-bandwidth
// bound (~6.5 us at 23.3 TB/s), with VALU a near co-bottleneck. Strategy:
// one wave per 16x16 tile; the affine projection field
//   t(y,x) = A*y + B*x + C
// is produced for the whole tile by a single V_WMMA_F32_16X16X4_F32
// (rank-2 outer product), and a lean clamp / residual / raw-v_sqrt epilogue
// writes non-temporal b32 stores sharing one base address (24-bit IOFFSET
// covers all 8 row offsets).

typedef __attribute__((ext_vector_type(2))) float v2f;
typedef __attribute__((ext_vector_type(8))) float v8f;

#define CANVAS 6144
#define LINEW  3.0f

__global__ __launch_bounds__(256) void path_raster2d_wmma(
    const float* __restrict__ kp,   // [2,2] normalized key points (y0,x0,y1,x1)
    float* __restrict__ out)        // [6144,6144] fp32
{
    // ---- uniform segment parameters (scalar loads + uniform VALU) ----
    const float p0y = kp[0] * (float)CANVAS;
    const float p0x = kp[1] * (float)CANVAS;
    const float d0  = kp[2] * (float)CANVAS - p0y;
    const float d1  = kp[3] * (float)CANVAS - p0x;
    const float len_sq = d0 * d0 + d1 * d1;
    const float inv = 1.0f / len_sq;            // one accurate divide per thread
    const float Ac  = d0 * inv;                 // t = Ac*y + Bc*x + Cc
    const float Bc  = d1 * inv;
    const float Cc  = -(p0y * d0 + p0x * d1) * inv;

    // ---- tile assignment: one wave per 16x16 tile ----
    const int wave = threadIdx.x >> 5;          // 0..7 (8 waves / block)
    const int lane = threadIdx.x & 31;
    const int l16  = lane & 15;                 // M index (A) / N index (B/D)
    const int half = (lane >> 4) & 1;           // lane-half: D rows +8
    const int tx   = (blockIdx.x * 8 + wave) << 4;  // tile col origin (gridDim.x=48)
    const int ty   = blockIdx.y << 4;               // tile row origin (gridDim.y=384)

    // ---- build rank-2 operands for V_WMMA_F32_16X16X4_F32 ----
    // Row m of A (16x4):  [u_m, 1, u_m, 1]   with u_m = Ac*(ty+m) + Cc
    // Col n of B (4x16):  [1/2, w_n/2, 1/2, w_n/2]  with w_n = Bc*(tx+n)
    // dot = u_m + w_n = t_unclamped(row, col); robust to the (vgpr,half)->K
    // mapping since A and B values depend only on the vgpr slot and the two
    // operands share the same layout.
    const float u = fmaf(Ac, (float)(ty + l16), Cc);
    const float w = Bc * (float)(tx + l16);
    v2f a; a[0] = u;    a[1] = 1.0f;
    v2f b; b[0] = 0.5f; b[1] = 0.5f * w;
    v8f c = {};
    // (neg_a, A, neg_b, B, c_mod, C, reuse_a, reuse_b)
    v8f t8 = __builtin_amdgcn_wmma_f32_16x16x4_f32(
        false, a, false, b, (short)0, c, false, false);

    // ---- lean per-element epilogue ----
    // D layout: lane l, vgpr v -> element (row = v + 8*half, col = l&15).
    // One 64-bit base address per lane; the 8 stores use compile-time byte
    // offsets v*6144*4 (< 2^23, fits signed 24-bit IOFFSET).
    const int   row0 = ty + (half << 3);
    const int   col  = tx + l16;
    const float vx   = (float)col  - p0x;       // constant across the 8 rows
    const float vy0  = (float)row0 - p0y;
    float* __restrict__ base = out + ((size_t)row0 * CANVAS + col);

#pragma unroll
    for (int v = 0; v < 8; ++v) {
        float t = fminf(fmaxf(t8[v], 0.0f), 1.0f);   // v_max clamp modifier
        const float ry = fmaf(-t, d0, vy0 + (float)v);
        const float rx = fmaf(-t, d1, vx);
        const float dist = __builtin_amdgcn_sqrtf(fmaf(ry, ry, rx * rx));
        const float res  = fmaxf(LINEW - dist, 0.0f); // == where(dist<3, 3-dist, 0)
        __builtin_nontemporal_store(res, base + v * CANVAS);
    }
}

extern "C" void kernel_launch(void* const* d_in, const int* in_sizes, int n_in,
                              void* d_out, int out_size, void* d_ws, size_t ws_size,
                              hipStream_t stream) {
    (void)in_sizes; (void)n_in; (void)out_size; (void)d_ws; (void)ws_size;
    const float* kp = (const float*)d_in[0];   // [2,2] float32
    float* out = (float*)d_out;                // [6144*6144] float32

    // 384x384 tiles of 16x16; 8 waves (tiles) per 256-thread block.
    dim3 grid(48, 384, 1);
    dim3 block(256, 1, 1);
    path_raster2d_wmma<<<grid, block, 0, stream>>>(kp, out);
}